// DgaRawSequence_61186104099021
// MI455X (gfx1250) — compile-verified
//
#include <hip/hip_runtime.h>
#include <hip/hip_bf16.h>

#define BB 256
#define TT 8192
#define II 6
#define UU 64
#define MM 16
#define UNFOLDS 6
#define EPSV 1e-8f

// gfx1250 hardware tanh (new CDNA5 transcendental): sigmoid(x)=0.5*tanh(0.5x)+0.5
__device__ __forceinline__ float fast_tanh(float x) {
#if __has_builtin(__builtin_amdgcn_tanhf)
    return __builtin_amdgcn_tanhf(x);
#elif __has_builtin(__builtin_amdgcn_tanh_f32)
    return __builtin_amdgcn_tanh_f32(x);
#else
    float r;
    asm("v_tanh_f32 %0, %1" : "=v"(r) : "v"(x));
    return r;
#endif
}

// CDNA5 async global->LDS copy (ASYNCcnt-tracked), with sync fallback.
#if __has_builtin(__builtin_amdgcn_global_load_async_to_lds_b32)
#define LTC_HAS_ASYNC_LDS 1
typedef __attribute__((address_space(1))) int* ltc_gas_ptr;   // global AS
typedef __attribute__((address_space(3))) int* ltc_lds_ptr;   // LDS AS
#endif

__device__ __forceinline__ void ltc_load_x_to_lds(const float* __restrict__ g,
                                                  float* __restrict__ l) {
#if defined(LTC_HAS_ASYNC_LDS)
    __builtin_amdgcn_global_load_async_to_lds_b32(
        (ltc_gas_ptr)g, (ltc_lds_ptr)l, 0, 0);
#else
    *l = *g;   // synchronous fallback keeps identical LDS data flow
#endif
}

__device__ __forceinline__ void ltc_wait_async() {
#if defined(LTC_HAS_ASYNC_LDS)
#if __has_builtin(__builtin_amdgcn_s_wait_asynccnt)
    __builtin_amdgcn_s_wait_asynccnt(0);
#else
    asm volatile("s_wait_asynccnt 0x0" ::: "memory");
#endif
#endif
}

// One block (128 threads = 4 wave32) per batch element.
// tid = h*64 + j : thread owns output unit j; h in {0,1} owns half of the
// i-reduction (i in [h*32, h*32+32) for recurrent, [h*3, h*3+3) for sensory).
__global__ __launch_bounds__(128)
void DgaRawSequence_61186104099021_kernel(
    const float* __restrict__ x,        // [B,T,I]
    const float* __restrict__ gleak,    // [U]
    const float* __restrict__ vleak,    // [U]
    const float* __restrict__ cm,       // [U]
    const float* __restrict__ sigma,    // [U,U]
    const float* __restrict__ mu,       // [U,U]
    const float* __restrict__ w,        // [U,U]
    const float* __restrict__ erev,     // [U,U]
    const float* __restrict__ ssig,     // [I,U]
    const float* __restrict__ smu,      // [I,U]
    const float* __restrict__ sw,       // [I,U]
    const float* __restrict__ serev,    // [I,U]
    const float* __restrict__ iw_g,     // [I]
    const float* __restrict__ ib_g,     // [I]
    const float* __restrict__ ow_g,     // [M]
    const float* __restrict__ ob_g,     // [M]
    float* __restrict__ out)            // [B,T,M]
{
    // LDS: transformed params + state + half-reduction scratch (~55 KB)
    __shared__ float2 pAB[UU * UU];   // (0.5*sigma, -0.5*mu*sigma)      32 KB
    __shared__ float  pWe[UU * UU];   // 0.5*w*erev  (|we| == 0.5*w)     16 KB
    __shared__ float2 sAB[II * UU];   // input affine folded in           3 KB
    __shared__ float  sWe[II * UU];   //                                1.5 KB
    __shared__ float  vsh[UU];
    __shared__ float  psn[128], psd[128];  // sensory half-partials
    __shared__ float  pun[128], pud[128];  // unfold  half-partials
    __shared__ float  xs[2][8];            // double-buffered raw x (async-to-LDS)

    const int tid = threadIdx.x;
    const int j   = tid & 63;
    const int h   = tid >> 6;
    const int b   = blockIdx.x;

    // ---- build transformed parameter tables in LDS ----
    for (int idx = tid; idx < UU * UU; idx += 128) {
        float sg = sigma[idx];
        float2 p;
        p.x = 0.5f * sg;
        p.y = -0.5f * mu[idx] * sg;
        pAB[idx] = p;
        pWe[idx] = 0.5f * w[idx] * erev[idx];
    }
    for (int idx = tid; idx < II * UU; idx += 128) {
        int  i  = idx >> 6;                    // sensory input index
        float sg = ssig[idx];
        float A  = 0.5f * sg;
        float Bc = -0.5f * smu[idx] * sg;
        float2 p;
        p.x = A * iw_g[i];                     // fold xi = x*iw + ib into table
        p.y = __builtin_fmaf(ib_g[i], A, Bc);
        sAB[idx] = p;
        sWe[idx] = 0.5f * sw[idx] * serev[idx];
    }
    __syncthreads();

    // ---- per-j constant bases (h==0 only; held in registers) ----
    float baseN = 0.f, baseD = 0.f, cmtv = 0.f, owr = 0.f, obr = 0.f;
    if (h == 0) {
        float swe = 0.f, swh = 0.f;
        for (int i = 0; i < UU; ++i) {
            float we = pWe[i * UU + j];
            swe += we;
            swh += __builtin_fabsf(we);
        }
        for (int i = 0; i < II; ++i) {
            float we = sWe[i * UU + j];
            swe += we;
            swh += __builtin_fabsf(we);
        }
        cmtv  = cm[j] * ((float)UNFOLDS);                    // cm/(ELAPSED/UNFOLDS), ELAPSED=1
        baseN = __builtin_fmaf(gleak[j], vleak[j], swe);     // gleak*vleak + Σwe (+sens)
        baseD = gleak[j] + cmtv + swh + EPSV;                // gleak + cm_t + Σ|we| (+sens) + eps
        if (j < MM) { owr = ow_g[j]; obr = ob_g[j]; }
    }

    float v = 0.f;                                           // hidden state (h==0 threads)
    const float* __restrict__ xb   = x   + (size_t)b * TT * II;
    float*       __restrict__ outb = out + (size_t)b * TT * MM;

    // kick off async fetch of x[t=0] into buffer 0
    if (tid < II) ltc_load_x_to_lds(xb + tid, &xs[0][tid]);

    for (int t = 0; t < TT; ++t) {
        const int cur = t & 1;
        ltc_wait_async();          // retire this wave's outstanding async->LDS loads
        if (h == 0) vsh[j] = v;    // publish state (first read is after psn barrier)
        __syncthreads();           // xs[cur] + vsh now visible to all waves

        // overlap next step's x fetch with the 6 ODE unfolds
        if (tid < II && (t + 1) < TT)
            ltc_load_x_to_lds(xb + (size_t)(t + 1) * II + tid, &xs[cur ^ 1][tid]);
        if (tid == 0) {
            int tp = t + 24; if (tp >= TT) tp = TT - 1;
            __builtin_prefetch(xb + (size_t)tp * II, 0, 0);  // -> global_prefetch_b8
        }

        // sensory synapses (constant across unfolds); split I=6 as 3+3
        float sn = 0.f, sd = 0.f;
#pragma unroll
        for (int ii = 0; ii < 3; ++ii) {
            int i = h * 3 + ii;
            float2 p  = sAB[i * UU + j];
            float  we = sWe[i * UU + j];
            float  th = fast_tanh(__builtin_fmaf(xs[cur][i], p.x, p.y));
            sn = __builtin_fmaf(we, th, sn);
            sd = __builtin_fmaf(__builtin_fabsf(we), th, sd);
        }
        psn[tid] = sn; psd[tid] = sd;
        __syncthreads();

        float nT = 0.f, dT = 0.f;
        if (h == 0) {
            nT = baseN + psn[j] + psn[64 + j];
            dT = baseD + psd[j] + psd[64 + j];
        }

        // semi-implicit Euler unfolds
        for (int k = 0; k < UNFOLDS; ++k) {
            float an = 0.f, ad = 0.f;
            const int ibase = h * 32;
#pragma unroll
            for (int ii = 0; ii < 32; ++ii) {
                int i = ibase + ii;
                float2 p  = pAB[i * UU + j];
                float  we = pWe[i * UU + j];
                float  th = fast_tanh(__builtin_fmaf(vsh[i], p.x, p.y));  // v_tanh_f32
                an = __builtin_fmaf(we, th, an);
                ad = __builtin_fmaf(__builtin_fabsf(we), th, ad);         // |we| via src-abs modifier
            }
            pun[tid] = an; pud[tid] = ad;
            __syncthreads();
            if (h == 0) {
                float num = __builtin_fmaf(cmtv, v, nT + an + pun[64 + j]);
                float den = dT + ad + pud[64 + j];
                v = num / den;                                // eps already folded into baseD
                vsh[j] = v;
            }
            __syncthreads();
        }

        if (h == 0 && j < MM)
            outb[(size_t)t * MM + j] = __builtin_fmaf(v, owr, obr);
    }
}

extern "C" void kernel_launch(void* const* d_in, const int* in_sizes, int n_in,
                              void* d_out, int out_size, void* d_ws, size_t ws_size,
                              hipStream_t stream) {
    (void)in_sizes; (void)n_in; (void)d_ws; (void)ws_size; (void)out_size;
    const float* xp    = (const float*)d_in[0];
    const float* gleak = (const float*)d_in[1];
    const float* vleak = (const float*)d_in[2];
    const float* cmv   = (const float*)d_in[3];
    const float* sigma = (const float*)d_in[4];
    const float* mu    = (const float*)d_in[5];
    const float* wv    = (const float*)d_in[6];
    const float* erev  = (const float*)d_in[7];
    const float* ssig  = (const float*)d_in[8];
    const float* smu   = (const float*)d_in[9];
    const float* swv   = (const float*)d_in[10];
    const float* serev = (const float*)d_in[11];
    const float* iw    = (const float*)d_in[12];
    const float* ib    = (const float*)d_in[13];
    const float* ow    = (const float*)d_in[14];
    const float* ob    = (const float*)d_in[15];
    float* outp = (float*)d_out;

    DgaRawSequence_61186104099021_kernel<<<BB, 128, 0, stream>>>(
        xp, gleak, vleak, cmv, sigma, mu, wv, erev,
        ssig, smu, swv, serev, iw, ib, ow, ob, outp);
}